// self_attention_76416058130984
// MI455X (gfx1250) — compile-verified
//
#include <hip/hip_runtime.h>
#include <hip/hip_bf16.h>
#include <stdint.h>

// Problem geometry (from reference): frames [4, 8, 64, 160, 160] fp32,
// out [8, 256, 160, 160] fp32.
constexpr int C            = 64;
constexpr int HW           = 160 * 160;        // 25600
constexpr int BATCH        = 8;
constexpr int FRAME_STRIDE = BATCH * C * HW;   // 13,107,200 floats
constexpr int BSTRIDE_IN   = C * HW;           // 1,638,400
constexpr int BSTRIDE_OUT  = 4 * C * HW;       // 6,553,600
constexpr int PIX_PER_BLK  = 256;
constexpr int CHUNK_C      = 8;                // channels per pipeline stage
constexpr int NCHUNK       = C / CHUNK_C;      // 8
constexpr int ROWS         = 4 * CHUNK_C;      // 32 rows (f-major) per chunk
constexpr int CHUNK_FLOATS = ROWS * PIX_PER_BLK; // 8192 floats = 32 KB

#define LOG2E 1.4426950408889634f

// One async 16B-per-lane copy: CDNA5 GLOBAL_LOAD_ASYNC_TO_LDS_B128 (ASYNCcnt).
__device__ __forceinline__ void async_b128(uint32_t lds_byte, const float* g) {
    asm volatile("global_load_async_to_lds_b128 %0, %1, off"
                 :: "v"(lds_byte), "v"(g)
                 : "memory");
}

// Issue one 32KB chunk: rows = f*8+cc, each row = 256 consecutive floats
// (pixels hw0..hw0+255) of frames[f][b][c0+cc][.]. 8 instructions per wave,
// each lane moves 16B, each instruction covers 512B of contiguous memory.
__device__ __forceinline__ void issue_chunk(const float* __restrict__ fr,
                                            int gPixBase /* b*BSTRIDE_IN + hw0 */,
                                            int c0, uint32_t ldsBuf, int t) {
    const int unit  = t & 63;   // 16B unit within a 1KB row
    const int rbase = t >> 6;   // 0..3
#pragma unroll
    for (int kk = 0; kk < 8; ++kk) {
        const int row = kk * 4 + rbase;          // 0..31
        const int f   = row >> 3;
        const int cc  = row & 7;
        const float* g = fr + f * FRAME_STRIDE + gPixBase + (c0 + cc) * HW + unit * 4;
        const uint32_t l = ldsBuf + (uint32_t)((kk * 256 + t) * 16);
        async_b128(l, g);
    }
}

__global__ __launch_bounds__(PIX_PER_BLK) void
fused_frame_softmax_gate(const float* __restrict__ fr, float* __restrict__ out) {
    __shared__ __align__(16) float smem[2][CHUNK_FLOATS];   // 64 KB double buffer

    const int t   = threadIdx.x;
    const int blk = blockIdx.x;
    const int b   = blk / (HW / PIX_PER_BLK);               // 100 blocks per batch image
    const int hw0 = (blk % (HW / PIX_PER_BLK)) * PIX_PER_BLK;

    const int gPixBase = b * BSTRIDE_IN + hw0;
    const int oPix     = b * BSTRIDE_OUT + hw0 + t;

    const uint32_t smem_base = (uint32_t)(uintptr_t)&smem[0][0]; // low 32 bits = LDS offset
    const uint32_t bufOff[2] = { smem_base, smem_base + (uint32_t)CHUNK_FLOATS * 4u };

    float Z[4] = {0.f, 0.f, 0.f, 0.f};

    // Prime the pipeline: two chunks in flight (16 outstanding ASYNC ops max).
    issue_chunk(fr, gPixBase, 0 * CHUNK_C, bufOff[0], t);
    issue_chunk(fr, gPixBase, 1 * CHUNK_C, bufOff[1], t);

    for (int k = 0; k < NCHUNK; ++k) {
        // Chunk completions are in-order: <=8 outstanding means chunk k landed.
        if (k < NCHUNK - 1) asm volatile("s_wait_asynccnt 8" ::: "memory");
        else                asm volatile("s_wait_asynccnt 0" ::: "memory");
        __syncthreads();                       // publish chunk k LDS to all waves

        const int bi = k & 1;
#pragma unroll
        for (int cc = 0; cc < CHUNK_C; ++cc) {
            float v[4], u[4];
#pragma unroll
            for (int f = 0; f < 4; ++f) v[f] = smem[bi][(f * 8 + cc) * 256 + t];
#pragma unroll
            for (int f = 0; f < 4; ++f) u[f] = v[f] * LOG2E;

            // Symmetric logits: 10 unique exps instead of 16.
            float e[4][4];
#pragma unroll
            for (int i = 0; i < 4; ++i)
#pragma unroll
                for (int j = i; j < 4; ++j) {
                    const float ex = __builtin_amdgcn_exp2f(u[i] * v[j]);
                    e[i][j] = ex;
                    e[j][i] = ex;
                }

            const int c = k * CHUNK_C + cc;
#pragma unroll
            for (int i = 0; i < 4; ++i) {
                Z[i] += (e[i][0] + e[i][1]) + (e[i][2] + e[i][3]);
                const float n = e[i][0] * v[0] + e[i][1] * v[1]
                              + e[i][2] * v[2] + e[i][3] * v[3];
                out[oPix + (i * 64 + c) * HW] = n;   // unnormalized numerator
            }
        }
        __syncthreads();                       // all waves done reading buf (k&1)
        if (k + 2 < NCHUNK)
            issue_chunk(fr, gPixBase, (k + 2) * CHUNK_C, bufOff[k & 1], t);
    }

    // Epilogue: scale this thread's 256 outputs by 1/Z_i. The re-read hits
    // L2/WGP$ (just written by this same thread), so no extra HBM traffic.
    float r[4];
#pragma unroll
    for (int i = 0; i < 4; ++i) r[i] = 1.0f / Z[i];
#pragma unroll 4
    for (int c = 0; c < C; ++c) {
#pragma unroll
        for (int i = 0; i < 4; ++i) {
            const int idx = oPix + (i * 64 + c) * HW;
            out[idx] = out[idx] * r[i];
        }
    }
}

extern "C" void kernel_launch(void* const* d_in, const int* in_sizes, int n_in,
                              void* d_out, int out_size, void* d_ws, size_t ws_size,
                              hipStream_t stream) {
    (void)in_sizes; (void)n_in; (void)out_size; (void)d_ws; (void)ws_size;
    const float* fr = (const float*)d_in[0];
    float* out      = (float*)d_out;
    const dim3 grid(BATCH * (HW / PIX_PER_BLK));   // 800 blocks
    const dim3 block(PIX_PER_BLK);                 // 256 threads = 8 waves
    hipLaunchKernelGGL(fused_frame_softmax_gate, grid, block, 0, stream, fr, out);
}